// Attention_25975962206620
// MI455X (gfx1250) — compile-verified
//
#include <hip/hip_runtime.h>

// B=2, H=16, S=2048, D=64 attention producing (out, weights).
// Bandwidth-bound on the 537 MB weights write (~27us floor @ 23.3 TB/s HBM).
// fp32 WMMA (v_wmma_f32_16x16x4_f32) keeps exact fp32 math vs the reference.
// Full 16x2048 logit row block lives in LDS (132 KB of 320 KB/WGP) so softmax
// needs no recompute pass and no extra HBM traffic. Weights/out stores are
// non-temporal so the 537 MB stream does not evict the L2-resident K/V/mask.

typedef float v2f __attribute__((ext_vector_type(2)));
typedef float v8f __attribute__((ext_vector_type(8)));

#define B_   2
#define H_   16
#define S_   2048
#define D_   64
#define QT   16          // q rows per workgroup
#define LROW 2056        // padded LDS row stride in floats (2048 + 8, bank-spread)
#define NW   4           // waves per block (wave32)
#define NT   128         // threads per block

__global__ __launch_bounds__(NT)
void attn_f32_wmma_kernel(const float* __restrict__ q,
                          const float* __restrict__ k,
                          const float* __restrict__ vv,
                          const float* __restrict__ mask,
                          float* __restrict__ out,
                          float* __restrict__ wts) {
    extern __shared__ float smem[];
    float* lg       = smem;               // [QT][LROW] logits -> exp -> weights
    float* partials = smem + QT * LROW;   // [16][8] reduction scratch
    float* rowmax   = partials + 128;     // [16]
    float* rinv     = rowmax + 16;        // [16]

    const int tid  = threadIdx.x;
    const int lane = tid & 31;
    const int wv   = tid >> 5;            // wave id 0..3
    const int r    = lane & 15;           // row-in-tile / col-in-tile index
    const int half = lane >> 4;           // selects K pair (0: K0/K1, 1: K2/K3)

    const int bid   = blockIdx.x;         // B*H*(S/QT) = 4096 blocks
    const int qt    = bid & (S_ / QT - 1);
    const int bh    = bid >> 7;
    const int b     = bh >> 4;            // /H
    const int qbase = qt * QT;

    const size_t bhS = (size_t)bh * S_;

    // ---- Q tile A-fragments: 16 chunks of K=4 covering D=64 ----
    // A(16x4) layout: lane r holds M=r; VGPR0/1 hold K = 2*half / 2*half+1.
    v2f aq[16];
    {
        const float* Qrow = q + (bhS + qbase + r) * D_;
        #pragma unroll
        for (int c = 0; c < 16; ++c)
            aq[c] = *(const v2f*)(Qrow + 4 * c + 2 * half);
    }

    // ---- Phase A: logits = (Q K^T)/8 + mask*(-1e9), tiles via WMMA f32 ----
    for (int kt = wv; kt < S_ / 16; kt += NW) {
        const int kg0 = kt * 16;
        const float* Krow = k + (bhS + kg0 + r) * D_;   // B frag: lane r = key col
        v8f acc = {0.f, 0.f, 0.f, 0.f, 0.f, 0.f, 0.f, 0.f};
        #pragma unroll
        for (int c = 0; c < 16; ++c) {
            v2f bf = *(const v2f*)(Krow + 4 * c + 2 * half);
            acc = __builtin_amdgcn_wmma_f32_16x16x4_f32(
                false, aq[c], false, bf, (short)0, acc, false, false);
        }
        // C layout: m = j + 8*half, n = r
        const int kg = kg0 + r;
        #pragma unroll
        for (int j = 0; j < 8; ++j) {
            const int m  = j + 8 * half;
            const int qg = qbase + m;
            float lv = acc[j] * 0.125f
                     + mask[((size_t)b * S_ + qg) * S_ + kg] * (-1.0e9f);
            lg[m * LROW + kg] = lv;
        }
    }
    __syncthreads();

    // ---- Phase B: row max ----
    {
        const int row = tid >> 3, sub = tid & 7;
        float mx = -3.4e38f;
        for (int t = 0; t < S_ / 8; ++t)
            mx = fmaxf(mx, lg[row * LROW + sub + 8 * t]);
        partials[row * 8 + sub] = mx;
    }
    __syncthreads();
    if (tid < 16) {
        float mx = partials[tid * 8];
        #pragma unroll
        for (int s2 = 1; s2 < 8; ++s2) mx = fmaxf(mx, partials[tid * 8 + s2]);
        rowmax[tid] = mx;
    }
    __syncthreads();

    // ---- Phase C: exp(l - max) in place + row sum ----
    {
        const int row = tid >> 3, sub = tid & 7;
        const float rm = rowmax[row];
        float s = 0.f;
        for (int t = 0; t < S_ / 8; ++t) {
            const int idx = row * LROW + sub + 8 * t;
            const float e = __expf(lg[idx] - rm);
            lg[idx] = e;
            s += e;
        }
        partials[row * 8 + sub] = s;
    }
    __syncthreads();
    if (tid < 16) {
        float s = 0.f;
        #pragma unroll
        for (int s2 = 0; s2 < 8; ++s2) s += partials[tid * 8 + s2];
        rinv[tid] = 1.0f / s;
    }
    __syncthreads();

    // ---- Phase D: normalize in LDS + coalesced non-temporal weights write ----
    for (int idx = tid; idx < QT * S_; idx += NT) {
        const int row = idx >> 11;          // S_ = 2048
        const int col = idx & (S_ - 1);
        const float w = lg[row * LROW + col] * rinv[row];
        lg[row * LROW + col] = w;
        // Streaming 537 MB write-once output: bypass L2 residency (TH=NT)
        __builtin_nontemporal_store(w, &wts[(bhS + qbase + row) * S_ + col]);
    }
    __syncthreads();

    // ---- Phase E: out = W @ V, wave wv owns 16 output columns ----
    {
        const int nbase = wv * 16;
        v8f acc = {0.f, 0.f, 0.f, 0.f, 0.f, 0.f, 0.f, 0.f};
        for (int kc = 0; kc < S_; kc += 4) {
            // A frag from LDS: W[q=r][kc + 2*half (+1)]
            v2f a = *(const v2f*)&lg[r * LROW + kc + 2 * half];
            // B frag from global V: V[kc+2*half (+1)][nbase + r]
            const float* Vp = vv + (bhS + kc + 2 * half) * D_ + nbase + r;
            v2f bf;
            bf.x = Vp[0];
            bf.y = Vp[D_];
            acc = __builtin_amdgcn_wmma_f32_16x16x4_f32(
                false, a, false, bf, (short)0, acc, false, false);
        }
        #pragma unroll
        for (int j = 0; j < 8; ++j) {
            const int m = j + 8 * half;
            __builtin_nontemporal_store(
                acc[j], &out[(bhS + qbase + m) * D_ + nbase + r]);
        }
    }
}

extern "C" void kernel_launch(void* const* d_in, const int* in_sizes, int n_in,
                              void* d_out, int out_size, void* d_ws, size_t ws_size,
                              hipStream_t stream) {
    (void)in_sizes; (void)n_in; (void)out_size; (void)d_ws; (void)ws_size;
    const float* q    = (const float*)d_in[0];
    const float* k    = (const float*)d_in[1];
    const float* v    = (const float*)d_in[2];
    const float* mask = (const float*)d_in[3];

    float* out = (float*)d_out;                       // (out, weights) concatenated
    float* wts = out + (size_t)B_ * H_ * S_ * D_;

    const size_t shmem = (size_t)(QT * LROW + 128 + 16 + 16) * sizeof(float);
    const int grid = B_ * H_ * (S_ / QT);             // 4096 blocks
    attn_f32_wmma_kernel<<<grid, NT, shmem, stream>>>(q, k, v, mask, out, wts);
}